// SegmentationPostProcess_13262859010629
// MI455X (gfx1250) — compile-verified
//
#include <hip/hip_runtime.h>

typedef float v2f __attribute__((ext_vector_type(2)));
typedef float v8f __attribute__((ext_vector_type(8)));

#define H_IM   480
#define W_IM   640
#define NBOX   128
#define MMASK  28
#define MI     30          // padded mask size (28 + ring of zeros)
#define CHUNKW 160         // columns per workgroup chunk (4 chunks of 160 = 640)
#define CONFTH 0.75f

__global__ __launch_bounds__(256)
void seg_paste_wmma_kernel(const float* __restrict__ boxes5,
                           const int*   __restrict__ classes,
                           const float* __restrict__ rawm,
                           int*   __restrict__ masks_out,
                           float* __restrict__ fout)
{
    // Padded 30x30 mask (rows/cols 30,31 stay zero = K padding), stride 33 (bank-conflict free)
    __shared__ float P[32][33];
    // T transposed: Tt[n][k] = (P @ Wx^T)[k][chunk*160+n]; stride 34 floats
    //  -> rows 8B aligned (136B), 34n mod 64 distinct over n in [0,16): conflict-free ds_load_b64
    __shared__ float Tt[CHUNKW][34];

    const int b     = (int)blockIdx.x >> 2;
    const int chunk = (int)blockIdx.x & 3;
    const int tid   = (int)threadIdx.x;
    const int lane  = tid & 31;
    const int wave  = tid >> 5;

    // ---------------- per-box scalars (redundant per thread, cheap) ----------------
    const float x0b   = boxes5[b * 5 + 0];
    const float y0b   = boxes5[b * 5 + 1];
    const float x1b   = boxes5[b * 5 + 2];
    const float y1b   = boxes5[b * 5 + 3];
    const float score = boxes5[b * 5 + 4];
    const int   cls   = classes[b];

    const float scl = (float)MI / (float)(MI - 2);   // 30/28
    const float wh  = (x1b - x0b) * 0.5f * scl;
    const float hh  = (y1b - y0b) * 0.5f * scl;
    const float xc  = (x1b + x0b) * 0.5f;
    const float yc  = (y1b + y0b) * 0.5f;

    const int ei0 = (int)truncf(xc - wh);
    const int ei1 = (int)truncf(yc - hh);
    const int ei2 = (int)truncf(xc + wh);
    const int ei3 = (int)truncf(yc + hh);

    const float ih = (float)max(ei3 - ei1 + 1, 1);
    const float iw = (float)max(ei2 - ei0 + 1, 1);

    const int p0x = min(max(ei0, 0), W_IM);
    const int p0y = min(max(ei1, 0), H_IM);
    const int p1x = min(max(ei2 + 1, 0), W_IM);
    const int p1y = min(max(ei3 + 1, 0), H_IM);

    const bool keep = (score > CONFTH) && (cls == 0);

    // pass-through outputs (boxes then scores after the mask block)
    if (chunk == 0) {
        if (tid < 4) fout[NBOX * H_IM * W_IM + b * 4 + tid] = boxes5[b * 5 + tid];
        if (tid == 4) fout[NBOX * H_IM * W_IM + NBOX * 4 + b] = score;
    }

    // ---------------- stage 0: build padded mask in LDS ----------------
    for (int t = tid; t < 32 * 33; t += 256) ((float*)P)[t] = 0.0f;
    __syncthreads();
    for (int t = tid; t < MMASK * MMASK; t += 256) {
        const int rr = t / MMASK;
        const int cc = t - rr * MMASK;
        P[rr + 1][cc + 1] = rawm[b * (MMASK * MMASK) + t];
    }
    __syncthreads();

    // ---------------- stage 1: T = P @ Wx^T  (Wx is 2-sparse: 2 FMAs/entry) ----------------
    const float invW = (float)MI / iw;
    for (int idx = tid; idx < CHUNKW * 32; idx += 256) {
        const int i = idx & 31;           // K row of T (contraction index for stage 2)
        const int n = idx >> 5;           // column within chunk
        const float dx  = (float)(chunk * CHUNKW + n - ei0);
        const float sx  = fminf(fmaxf((dx + 0.5f) * invW - 0.5f, 0.0f), (float)(MI - 1));
        const float fx0 = floorf(sx);
        const int   kx0 = (int)fx0;
        const float wx  = sx - fx0;
        const int   kx1 = min(kx0 + 1, MI - 1);
        // rows i>=30 of P are zero -> K padding comes out as 0 automatically
        Tt[n][i] = P[i][kx0] * (1.0f - wx) + P[i][kx1] * wx;
    }
    __syncthreads();

    // ---------------- stage 2: out = Wy @ T via V_WMMA_F32_16X16X4_F32 ----------------
    const float invH = (float)MI / ih;
    const int   kOff = (lane < 16) ? 0 : 2;   // A/B K-pair held by this lane half

    for (int mt = wave; mt < H_IM / 16; mt += 8) {
        // A fragment (Wy rows, 2-sparse) synthesized in registers, reused across x-tiles.
        // 32-bit A 16x4 layout: lanes 0-15 & 16-31 both M=lane&15; VGPR0=K{0|2}, VGPR1=K{1|3}.
        const int   mrow = mt * 16 + (lane & 15);
        const float dyv  = (float)(mrow - ei1);
        const float syv  = fminf(fmaxf((dyv + 0.5f) * invH - 0.5f, 0.0f), (float)(MI - 1));
        const float fy0  = floorf(syv);
        const int   ky0  = (int)fy0;
        const float wy   = syv - fy0;
        const int   ky1  = min(ky0 + 1, MI - 1);

        v2f a[8];
#pragma unroll
        for (int s = 0; s < 8; ++s) {
            const int k0 = 4 * s + kOff;
            const int k1 = k0 + 1;
            a[s].x = ((k0 == ky0) ? (1.0f - wy) : 0.0f) + ((k0 == ky1) ? wy : 0.0f);
            a[s].y = ((k1 == ky0) ? (1.0f - wy) : 0.0f) + ((k1 == ky1) ? wy : 0.0f);
        }

        for (int xt = 0; xt < CHUNKW / 16; ++xt) {
            v8f c = {};
#pragma unroll
            for (int s = 0; s < 8; ++s) {
                // B fragment 4x16: lane holds N=lane&15, K pair {kOff, kOff+1} at base 4s.
                const v2f bfrag = *(const v2f*)&Tt[xt * 16 + (lane & 15)][4 * s + kOff];
                c = __builtin_amdgcn_wmma_f32_16x16x4_f32(
                        /*neg_a=*/false, a[s], /*neg_b=*/false, bfrag,
                        /*c_mod=*/(short)0, c, /*reuse_a=*/false, /*reuse_b=*/false);
            }

            // epilogue: threshold + inside-box + keep, int32 store
            const int  xcol = chunk * CHUNKW + xt * 16 + (lane & 15);
            const bool xin  = (xcol >= p0x) && (xcol < p1x);
            const int  base = b * (H_IM * W_IM) + xcol;
#pragma unroll
            for (int v = 0; v < 8; ++v) {
                // C layout: VGPR v -> M=v (lanes 0-15) / M=v+8 (lanes 16-31), N=lane&15
                const int  yrow = mt * 16 + v + ((lane >= 16) ? 8 : 0);
                const bool on   = keep && xin && (yrow >= p0y) && (yrow < p1y) && (c[v] > 0.5f);
                masks_out[base + yrow * W_IM] = on ? 1 : 0;
            }
        }
    }
}

extern "C" void kernel_launch(void* const* d_in, const int* in_sizes, int n_in,
                              void* d_out, int out_size, void* d_ws, size_t ws_size,
                              hipStream_t stream) {
    (void)in_sizes; (void)n_in; (void)d_ws; (void)ws_size; (void)out_size;
    const float* boxes5  = (const float*)d_in[0];   // [1,128,5] f32
    const int*   classes = (const int*)  d_in[1];   // [1,128] int
    const float* rawm    = (const float*)d_in[2];   // [1,128,28,28] f32
    int*   masks_out = (int*)d_out;                 // [128,480,640] int32
    float* fout      = (float*)d_out;               // boxes/scores tail (float32)

    // 128 boxes x 4 column-chunks; 256 threads (8 wave32) per workgroup
    seg_paste_wmma_kernel<<<dim3(NBOX * 4), dim3(256), 0, stream>>>(
        boxes5, classes, rawm, masks_out, fout);
}